// GCN_56813827392376
// MI455X (gfx1250) — compile-verified
//
#include <hip/hip_runtime.h>
#include <hip/hip_bf16.h>

typedef float v2f __attribute__((ext_vector_type(2)));
typedef float v8f __attribute__((ext_vector_type(8)));

#define CH 128          // hidden width (all GEMMs have 128 output cols)
#define NGRAPH 64
#define ASTRIDE 20      // LDS row stride (dwords) -> conflict-free b64 frag loads

// Native fp32 atomic add (no return): device scope, performed at the shared L2.
__device__ __forceinline__ void fatomicAdd(float* p, float v) {
    asm volatile("global_atomic_add_f32 %0, %1, off scope:SCOPE_DEV"
                 :: "v"(p), "v"(v) : "memory");
}

// CDNA5 async copy: 16B global -> 16B LDS per lane (EXEC-masked), ASYNCcnt-tracked.
__device__ __forceinline__ void async_copy16(unsigned lds_off, const float* gptr) {
    asm volatile("global_load_async_to_lds_b128 %0, %1, off"
                 :: "v"(lds_off), "v"(gptr) : "memory");
}
__device__ __forceinline__ void wait_async0() {
    asm volatile("s_wait_asynccnt 0x0" ::: "memory");
}

// ---------------------------------------------------------------- utilities
__global__ void zero_f32(float* __restrict__ p, long n) {
    long t = (long)blockIdx.x * blockDim.x + threadIdx.x;
    long stride = (long)gridDim.x * blockDim.x;
    for (; t < n; t += stride) p[t] = 0.0f;
}

__global__ void deg_kernel(const int* __restrict__ src, const int* __restrict__ dst,
                           float* __restrict__ outdeg, float* __restrict__ indeg, int E) {
    int t = blockIdx.x * blockDim.x + threadIdx.x;
    if (t < E) {
        fatomicAdd(&outdeg[src[t]], 1.0f);
        fatomicAdd(&indeg[dst[t]], 1.0f);
    }
}

__global__ void invsqrt_kernel(float* __restrict__ p, long n) {
    long t = (long)blockIdx.x * blockDim.x + threadIdx.x;
    if (t < n) p[t] = rsqrtf(fmaxf(p[t], 1.0f));
}

// ------------------------------------------------ fp32 WMMA GEMM, 128 cols
// C[Nrows,128] = relu?( rowScale[r] * ([A1|A2] @ W) + bias ) (+ residual)
// (rowScale applied in epilog: diag(s)AW == diag(s)(AW))
__launch_bounds__(256)
__global__ void gemm_wmma_f32(const float* __restrict__ A1, int lda1, int K1,
                              const float* __restrict__ A2, int lda2, int K2,
                              const float* __restrict__ Wm,
                              const float* __restrict__ bias,
                              const float* __restrict__ rowScale,
                              const float* __restrict__ residual,
                              float* __restrict__ C, int Nrows, int doReluRes) {
    __shared__ float As[128 * ASTRIDE];
    __shared__ float Bs[128 * ASTRIDE];   // transposed: Bs[n*ASTRIDE + k]

    const int t    = threadIdx.x;
    const int w    = t >> 5;        // wave 0..7 -> 16-row stripe
    const int lane = t & 31;
    const int rowBlk = blockIdx.x * 128;
    const int Ktot = K1 + K2;

    // Tail block: pre-zero the A tile; async lanes for invalid rows stay EXEC-off,
    // so those LDS rows remain zero for every K chunk.
    if (rowBlk + 128 > Nrows) {
        for (int i = t; i < 128 * ASTRIDE; i += 256) As[i] = 0.0f;
    }

    v8f acc[8] = {};

    const int m     = lane & 15;          // M (or N) index inside fragment
    const int khalf = (lane >> 4) * 2;    // K pair selector (ISA 16x4 f32 layout)

    // Per-thread A staging geometry: half row (8 floats = two b128 copies)
    const int ar  = t >> 1;
    const int akk = (t & 1) * 8;
    const int aR  = rowBlk + ar;
    const unsigned aLds0 = (unsigned)(uintptr_t)&As[ar * ASTRIDE + akk];
    const unsigned aLds1 = aLds0 + 16u;

    for (int k0 = 0; k0 < Ktot; k0 += 16) {
        const float* Asrc; int lda; int kc;
        if (k0 < K1) { Asrc = A1; lda = lda1; kc = k0; }
        else         { Asrc = A2; lda = lda2; kc = k0 - K1; }

        // ---- stage A chunk [128 rows x 16 k] via async global->LDS b128
        if (aR < Nrows) {
            const float* p = Asrc + (size_t)aR * lda + kc + akk;
            async_copy16(aLds0, p);
            async_copy16(aLds1, p + 4);
        }
        // ---- stage B chunk [16 k x 128 n], transposed into LDS
        {
            const float* Wp = Wm + (size_t)k0 * CH;
            #pragma unroll
            for (int i = 0; i < 8; i++) {
                int e = t + 256 * i;
                int k = e >> 7, n = e & 127;
                Bs[n * ASTRIDE + k] = Wp[k * CH + n];
            }
        }
        wait_async0();          // our async copies landed in LDS
        __syncthreads();        // + everyone's ds stores / async copies visible

        #pragma unroll
        for (int kk = 0; kk < 4; kk++) {
            v2f a = *(const v2f*)&As[(w * 16 + m) * ASTRIDE + kk * 4 + khalf];
            #pragma unroll
            for (int nt = 0; nt < 8; nt++) {
                v2f b = *(const v2f*)&Bs[(nt * 16 + m) * ASTRIDE + kk * 4 + khalf];
                acc[nt] = __builtin_amdgcn_wmma_f32_16x16x4_f32(
                    false, a, false, b, (short)0, acc[nt], false, false);
            }
        }
        __syncthreads();
    }

    // ---- epilog: C layout (VGPR v: lanes 0-15 -> M=v, lanes 16-31 -> M=v+8)
    const int rofs = (lane >> 4) * 8;
    float rs_v[8];
    #pragma unroll
    for (int v = 0; v < 8; v++) {
        int row = rowBlk + w * 16 + v + rofs;
        rs_v[v] = (rowScale && row < Nrows) ? rowScale[row] : 1.0f;
    }
    #pragma unroll
    for (int nt = 0; nt < 8; nt++) {
        int col = nt * 16 + m;
        float bv = bias ? bias[col] : 0.0f;
        #pragma unroll
        for (int v = 0; v < 8; v++) {
            int row = rowBlk + w * 16 + v + rofs;
            if (row < Nrows) {
                float x = acc[nt][v] * rs_v[v] + bv;
                if (doReluRes) x = fmaxf(x, 0.0f) + residual[(size_t)row * CH + col];
                C[(size_t)row * CH + col] = x;
            }
        }
    }
}

// ---------------------------------------------------------------- BatchNorm
__global__ void bn_stats(const float* __restrict__ h, int N,
                         float* __restrict__ sums, float* __restrict__ sumsq) {
    int c = threadIdx.x;  // 128 threads
    float s = 0.f, s2 = 0.f;
    for (int r = blockIdx.x; r < N; r += gridDim.x) {
        float x = h[(size_t)r * CH + c];
        s += x; s2 += x * x;
    }
    fatomicAdd(&sums[c], s);
    fatomicAdd(&sumsq[c], s2);
}

__global__ void bn_finalize(const float* __restrict__ sums, const float* __restrict__ sumsq,
                            const float* __restrict__ gamma, const float* __restrict__ beta,
                            float* __restrict__ scale, float* __restrict__ shift, int N) {
    int c = threadIdx.x;  // 128 threads
    float invn = 1.0f / (float)N;
    float mean = sums[c] * invn;
    float var  = sumsq[c] * invn - mean * mean;
    float sc   = gamma[c] * rsqrtf(var + 1e-5f);
    scale[c] = sc;
    shift[c] = beta[c] - mean * sc;
}

// hs = bn(h) * out_inv_sqrt (row scale), and zero the agg buffer
__global__ void stage_kernel(const float* __restrict__ h,
                             const float* __restrict__ scale, const float* __restrict__ shift,
                             const float* __restrict__ oinv,
                             float* __restrict__ hs, float* __restrict__ agg, long total) {
    long t = (long)blockIdx.x * blockDim.x + threadIdx.x;
    if (t < total) {
        long i = t >> 7; int c = (int)(t & 127);
        hs[t]  = (h[t] * scale[c] + shift[c]) * oinv[i];
        agg[t] = 0.0f;
    }
}

// agg[dst] += hs[src]  (32 lanes/edge, 4 channels/lane: 512B coalesced gather)
__global__ void scatter_kernel(const float* __restrict__ hs,
                               const int* __restrict__ src, const int* __restrict__ dst,
                               float* __restrict__ agg, int E) {
    long t = (long)blockIdx.x * blockDim.x + threadIdx.x;
    long e = t >> 5;
    if (e >= E) return;
    int c4 = (int)(t & 31) * 4;
    int s = src[e], d = dst[e];
    float4 v = *(const float4*)&hs[(size_t)s * CH + c4];
    float* p = &agg[(size_t)d * CH + c4];
    fatomicAdd(p + 0, v.x);
    fatomicAdd(p + 1, v.y);
    fatomicAdd(p + 2, v.z);
    fatomicAdd(p + 3, v.w);
}

// ---------------------------------------------------------------- pooling
__global__ void pool_accum(const float* __restrict__ h, const int* __restrict__ gid,
                           float* __restrict__ gsum, float* __restrict__ gcnt, long total) {
    long t = (long)blockIdx.x * blockDim.x + threadIdx.x;
    if (t < total) {
        long i = t >> 7; int c = (int)(t & 127);
        int g = gid[i];
        fatomicAdd(&gsum[(size_t)g * CH + c], h[t]);
        if (c == 0) fatomicAdd(&gcnt[g], 1.0f);
    }
}

__global__ void pool_final(const float* __restrict__ gsum, const float* __restrict__ gcnt,
                           float* __restrict__ out) {
    int t = blockIdx.x * blockDim.x + threadIdx.x;  // 64*128
    if (t < NGRAPH * CH) {
        int g = t >> 7;
        out[t] = gsum[t] / fmaxf(gcnt[g], 1.0f);
    }
}

// ---------------------------------------------------------------- launcher
extern "C" void kernel_launch(void* const* d_in, const int* in_sizes, int n_in,
                              void* d_out, int out_size, void* d_ws, size_t ws_size,
                              hipStream_t stream) {
    const float* h2  = (const float*)d_in[0];
    const float* h3  = (const float*)d_in[1];
    const float* W1  = (const float*)d_in[2];
    const float* b1  = (const float*)d_in[3];
    const float* W2  = (const float*)d_in[4];
    const float* b2  = (const float*)d_in[5];
    const float* g1  = (const float*)d_in[6];
    const float* be1 = (const float*)d_in[7];
    const float* g2  = (const float*)d_in[8];
    const float* be2 = (const float*)d_in[9];
    const float* cW1 = (const float*)d_in[10];
    const float* cb1 = (const float*)d_in[11];
    const float* cW2 = (const float*)d_in[12];
    const float* cb2 = (const float*)d_in[13];
    const int*   src = (const int*)d_in[14];
    const int*   dst = (const int*)d_in[15];
    const int*   gid = (const int*)d_in[16];

    const int N = in_sizes[0] / 256;   // IN2 = 256
    const int E = in_sizes[14];
    const long NB = (long)N * CH;

    float* ws     = (float*)d_ws;
    float* buf0   = ws;                    // h2p, later hs (scaled bn output)
    float* buf1   = buf0 + NB;             // h (persistent, residual)
    float* buf2   = buf1 + NB;             // agg (scatter destination)
    float* oinv   = buf2 + NB;             // out-degree^{-1/2}
    float* iinv   = oinv + N;              // in-degree^{-1/2}  (contiguous with oinv)
    float* bsum   = iinv + N;
    float* bsq    = bsum + CH;
    float* bscale = bsq + CH;
    float* bshift = bscale + CH;
    float* gsum   = bshift + CH;           // [64,128]
    float* gcnt   = gsum + NGRAPH * CH;    // [64]   (contiguous with gsum)

    // ---- degrees -> inv sqrt
    zero_f32<<<512, 256, 0, stream>>>(oinv, 2L * N);
    deg_kernel<<<(E + 255) / 256, 256, 0, stream>>>(src, dst, oinv, iinv, E);
    invsqrt_kernel<<<(2 * N + 255) / 256, 256, 0, stream>>>(oinv, 2L * N);

    const int gblocks = (N + 127) / 128;
    // h2p = h2 @ W1 + b1
    gemm_wmma_f32<<<gblocks, 256, 0, stream>>>(h2, 256, 256, nullptr, 0, 0,
                                               W1, b1, nullptr, nullptr, buf0, N, 0);
    // h = [h2p | h3] @ W2 + b2   (concat never materialized)
    gemm_wmma_f32<<<gblocks, 256, 0, stream>>>(buf0, CH, CH, h3, CH, CH,
                                               W2, b2, nullptr, nullptr, buf1, N, 0);

    const long tot = NB;
    const int nodeGrid = (int)((tot + 255) / 256);
    const int edgeGrid = (int)(((long)E * 32 + 255) / 256);

    for (int layer = 0; layer < 2; layer++) {
        const float* gam = layer ? g2 : g1;
        const float* bet = layer ? be2 : be1;
        const float* cW  = layer ? cW2 : cW1;
        const float* cb  = layer ? cb2 : cb1;

        zero_f32<<<1, 256, 0, stream>>>(bsum, 2 * CH);
        bn_stats<<<512, CH, 0, stream>>>(buf1, N, bsum, bsq);
        bn_finalize<<<1, CH, 0, stream>>>(bsum, bsq, gam, bet, bscale, bshift, N);
        // hs = bn(h) * D_out^{-1/2}; agg = 0
        stage_kernel<<<nodeGrid, 256, 0, stream>>>(buf1, bscale, bshift, oinv, buf0, buf2, tot);
        // agg[dst] += hs[src]
        scatter_kernel<<<edgeGrid, 256, 0, stream>>>(buf0, src, dst, buf2, E);
        // h = h + relu( (agg @ cW) * D_in^{-1/2} + cb )
        gemm_wmma_f32<<<gblocks, 256, 0, stream>>>(buf2, CH, CH, nullptr, 0, 0,
                                                   cW, cb, iinv, buf1, buf1, N, 1);
    }

    // ---- per-graph mean pooling
    zero_f32<<<1, 256, 0, stream>>>(gsum, NGRAPH * CH + NGRAPH);
    pool_accum<<<nodeGrid, 256, 0, stream>>>(buf1, gid, gsum, gcnt, tot);
    pool_final<<<(NGRAPH * CH + 255) / 256, 256, 0, stream>>>(gsum, gcnt, (float*)d_out);
}